// GlobalConditionedBias_67645734912846
// MI455X (gfx1250) — compile-verified
//
#include <hip/hip_runtime.h>
#include <hip/hip_bf16.h>
#include <math.h>

#define B_   4
#define T_   1024
#define HID_ 16
#define H_   16

typedef __attribute__((ext_vector_type(2))) float v2f;
typedef __attribute__((ext_vector_type(4))) float v4f;
typedef __attribute__((ext_vector_type(8))) float v8f;

// ---------------------------------------------------------------------------
// Pre-pass: sin/cos of dphi for every (b, t).  B*T = 4096 elements, 32 KB ws.
// ---------------------------------------------------------------------------
__global__ __launch_bounds__(256) void gcb_prep_kernel(
    const float* __restrict__ globals_,
    const float* __restrict__ tokens,   // (B, T, 4)
    float* __restrict__ sinb,           // (B*T)
    float* __restrict__ cosb)           // (B*T)
{
    int idx = blockIdx.x * blockDim.x + threadIdx.x;
    if (idx >= B_ * T_) return;
    int b = idx >> 10;                      // idx / T_
    float phi  = tokens[idx * 4 + 3];
    float dphi = phi - globals_[b * 2 + 1]; // met_phi
    sinb[idx] = sinf(dphi);
    cosb[idx] = cosf(dphi);
}

// ---------------------------------------------------------------------------
// Main: one wave = one 16x16 (i,j) tile of one batch.
//   layer1: h[i,j,k] = relu(u[i,k] + v'[j,k])        (rank-structured, VALU)
//   layer2: 16x16x16 f32 matmul via 4x v_wmma_f32_16x16x4_f32 (K=16)
//           tanh(gate) folded into W2; acc initialized with tg*b2.
//   D rows = j, D cols = channel  ->  contiguous-j NT stores.
// ---------------------------------------------------------------------------
__global__ __launch_bounds__(256) void gcb_bias_kernel(
    const float* __restrict__ globals_,     // (B,2)
    const unsigned char* __restrict__ mask, // (B,T) bool
    const float* __restrict__ W1,           // (5,16)
    const float* __restrict__ b1,           // (16)
    const float* __restrict__ W2,           // (16,16)
    const float* __restrict__ b2,           // (16)
    const float* __restrict__ gate,         // (1)
    const float* __restrict__ sinb,         // (B*T)
    const float* __restrict__ cosb,         // (B*T)
    float* __restrict__ out)                // (B,16,T,T)
{
    const int lane = threadIdx.x & 31;
    const int wave = threadIdx.x >> 5;
    const int tile = blockIdx.x * 8 + wave;   // 16384 tiles total

    const int b   = tile >> 12;          // / (64*64)
    const int rem = tile & 4095;
    const int i0  = (rem >> 6) << 4;     // i-tile * 16
    const int j0  = (rem & 63) << 4;     // j-tile * 16

    const int m  = lane & 15;            // A-row (=j within tile) / B-col (=channel)
    const int hi = lane >> 4;            // lane half selects K offset {0,1} vs {2,3}

    const float mag = globals_[b * 2 + 0];
    const float tg  = tanhf(gate[0]);

    // --- per-lane layer-1 column features (j = j0 + m, same for both halves)
    const int j = j0 + m;
    const float sj = sinb[b * T_ + j];
    const float cj = cosb[b * T_ + j];

    // lane's K set for WMMA q: k = 4q + 2*hi + {0,1}
    float w10[8], w11[8], vp[8];
#pragma unroll
    for (int q = 0; q < 4; ++q) {
#pragma unroll
        for (int t = 0; t < 2; ++t) {
            const int k = 4 * q + 2 * hi + t;
            w10[2 * q + t] = W1[0 * 16 + k];
            w11[2 * q + t] = W1[1 * 16 + k];
            vp [2 * q + t] = sj * W1[2 * 16 + k] + cj * W1[3 * 16 + k]
                           + mag * W1[4 * 16 + k] + b1[k];
        }
    }

    // --- (tg * W2) resident in B-operand layout: V0 = W2[k0,n], V1 = W2[k0+1,n]
    v2f bmat[4];
#pragma unroll
    for (int q = 0; q < 4; ++q) {
        const int k0 = 4 * q + 2 * hi;
        bmat[q].x = tg * W2[(k0 + 0) * 16 + m];
        bmat[q].y = tg * W2[(k0 + 1) * 16 + m];
    }

    const float tb2 = tg * b2[m];        // folded bias (accumulator init value)

    // mask over the 8 j-rows this lane's D registers cover (j0 + 8*hi + r)
    float mj[8];
#pragma unroll
    for (int r = 0; r < 8; ++r)
        mj[r] = mask[b * T_ + j0 + 8 * hi + r] ? 1.0f : 0.0f;

    // i-tile mask: 16 bytes -> 16-bit register bitmask (no loads in hot loop)
    unsigned mbits = 0;
#pragma unroll
    for (int r = 0; r < 16; ++r)
        mbits |= (mask[b * T_ + i0 + r] ? 1u : 0u) << r;

    float* obase = out + (size_t)(b * 16 + m) * (size_t)(T_ * T_)
                       + (size_t)i0 * T_ + (size_t)(j0 + 8 * hi);

    for (int ii = 0; ii < 16; ++ii) {
        const int i  = i0 + ii;
        const float si = sinb[b * T_ + i];
        const float ci = cosb[b * T_ + i];

        v8f acc;
#pragma unroll
        for (int r = 0; r < 8; ++r) acc[r] = tb2;   // C starts at tg*b2[n]

#pragma unroll
        for (int q = 0; q < 4; ++q) {
            v2f a;
            a.x = fmaxf(si * w10[2 * q + 0] + ci * w11[2 * q + 0] + vp[2 * q + 0], 0.0f);
            a.y = fmaxf(si * w10[2 * q + 1] + ci * w11[2 * q + 1] + vp[2 * q + 1], 0.0f);
            // D = A(16x4 f32) x B(4x16 f32) + C   -- accumulate K=16 in 4 steps
            acc = __builtin_amdgcn_wmma_f32_16x16x4_f32(
                false, a, false, bmat[q], (short)0, acc, false, false);
        }

        const float gi = ((mbits >> ii) & 1u) ? 1.0f : 0.0f;   // mask[b,i]
        v4f o0, o1;
        o0.x = gi * mj[0] * acc[0];
        o0.y = gi * mj[1] * acc[1];
        o0.z = gi * mj[2] * acc[2];
        o0.w = gi * mj[3] * acc[3];
        o1.x = gi * mj[4] * acc[4];
        o1.y = gi * mj[5] * acc[5];
        o1.z = gi * mj[6] * acc[6];
        o1.w = gi * mj[7] * acc[7];

        v4f* p = (v4f*)(obase + (size_t)ii * T_);
        __builtin_nontemporal_store(o0, p + 0);   // output never re-read: bypass caches
        __builtin_nontemporal_store(o1, p + 1);
    }
}

// ---------------------------------------------------------------------------
extern "C" void kernel_launch(void* const* d_in, const int* in_sizes, int n_in,
                              void* d_out, int out_size, void* d_ws, size_t ws_size,
                              hipStream_t stream) {
    const float* globals_ = (const float*)d_in[0];
    const float* tokens   = (const float*)d_in[1];
    const unsigned char* mask = (const unsigned char*)d_in[2];
    const float* W1   = (const float*)d_in[3];
    const float* b1   = (const float*)d_in[4];
    const float* W2   = (const float*)d_in[5];
    const float* b2   = (const float*)d_in[6];
    const float* gate = (const float*)d_in[7];
    float* out = (float*)d_out;

    float* sinb = (float*)d_ws;              // B*T floats
    float* cosb = sinb + B_ * T_;            // B*T floats (32 KB total)

    gcb_prep_kernel<<<(B_ * T_ + 255) / 256, 256, 0, stream>>>(
        globals_, tokens, sinb, cosb);

    // 16384 tiles, 8 waves (tiles) per 256-thread block
    gcb_bias_kernel<<<2048, 256, 0, stream>>>(
        globals_, mask, W1, b1, W2, b2, gate, sinb, cosb, out);
}